// nms1_78907139162556
// MI455X (gfx1250) — compile-verified
//
#include <hip/hip_runtime.h>
#include <stdint.h>

#define NUM_CLASSES 80
#define NBOX        25200
#define BATCH       32
#define MAX_B       100
#define IOU_T       0.5f
#define SCORE_T     0.25f
#define NEG         (-1e9f)

#define BLOCK_NMS   1024
#define KGROUPS     7                  // quads per thread
#define NPAD        (4 * BLOCK_NMS * KGROUPS)   // 28672 floats in LDS
#define SCORE_BYTES (NBOX * 4)         // 100800, multiple of 16

typedef float    v4f __attribute__((ext_vector_type(4)));
typedef uint32_t v4u __attribute__((ext_vector_type(4)));
typedef int      v4i __attribute__((ext_vector_type(4)));
typedef int      v8i __attribute__((ext_vector_type(8)));

__device__ __forceinline__ float clamp01(float v) {
  return fminf(fmaxf(v, 0.0f), 1.0f);
}

// ---------------------------------------------------------------------------
// Kernel A: fused class-argmax + score + threshold.  Streaming, HBM-bound
// (258 MB in -> 3.2 MB out, ~11 us roofline at 23.3 TB/s).  One thread per
// box, 20x b128 non-temporal loads so the one-shot logits don't evict the
// L2-resident boxes/scores needed by the NMS pass.
// ---------------------------------------------------------------------------
__global__ void score_kernel(const float* __restrict__ p20,
                             const float* __restrict__ c20,
                             float* __restrict__ scores,
                             int total) {
  int i = blockIdx.x * blockDim.x + threadIdx.x;
  if (i >= total) return;

  const v4f* cp = (const v4f*)(c20 + (size_t)i * NUM_CLASSES);
  float bv = -3.4e38f;
  int   bi = 0;
#pragma unroll
  for (int q = 0; q < NUM_CLASSES / 4; ++q) {
    v4f v = __builtin_nontemporal_load(cp + q);
    int base = q * 4;
    if (v.x > bv) { bv = v.x; bi = base + 0; }   // strict '>' => first index on ties
    if (v.y > bv) { bv = v.y; bi = base + 1; }
    if (v.z > bv) { bv = v.z; bi = base + 2; }
    if (v.w > bv) { bv = v.w; bi = base + 3; }
  }
  float s = p20[i] * (float)bi;                  // score = objectness * class index
  scores[i] = (s > SCORE_T) ? s : NEG;           // pre-threshold like reference sc0
}

// ---------------------------------------------------------------------------
// Per-element IoU suppression decision (divide-free: iou>T <=> inter>T*union).
// ---------------------------------------------------------------------------
__device__ __forceinline__ float supp_one(float cur, int idx, int si,
                                          v4f sb, float sArea, v4f bx) {
  float iy = fminf(sb.z, bx.z) - fmaxf(sb.x, bx.x);
  float ix = fminf(sb.w, bx.w) - fmaxf(sb.y, bx.y);
  iy = fmaxf(iy, 0.0f);
  ix = fmaxf(ix, 0.0f);
  float inter = iy * ix;
  float areaJ = (bx.z - bx.x) * (bx.w - bx.y);
  float uni   = sArea + areaJ - inter;
  bool  sup   = (uni > 0.0f) && (inter > IOU_T * uni);
  return (sup || idx == si) ? NEG : cur;
}

// ---------------------------------------------------------------------------
// Kernel B: greedy NMS, one workgroup (1024 thr = 32 waves) per batch.
//   - scores staged into LDS by the Tensor Data Mover (one instruction,
//     TENSORcnt-tracked); async-b128 DMA fallback if the builtin is absent
//   - per iteration: b128 LDS argmax scan -> wave32 shfl_xor reduce ->
//     cross-wave reduce; then b128 IoU suppression vs L2-resident boxes
// ---------------------------------------------------------------------------
__global__ __launch_bounds__(BLOCK_NMS) void nms_kernel(
    const float* __restrict__ bbox,       // [B, N, 4]  (y1,x1,y2,x2)
    const float* __restrict__ scores_in,  // [B, N]
    float* __restrict__ outPred,          // [B, MAX_B, 6]
    int* __restrict__ outValid)           // [B]
{
  extern __shared__ float smem[];
  float* s_sc  = smem;                    // NPAD floats (114688 B)
  float* r_val = smem + NPAD;             // 32 floats
  int*   r_idx = (int*)(r_val + 32);      // 32 ints
  float* bc    = (float*)(r_idx + 32);    // [0]=val, [1]=idx (bitcast)

  const int b = blockIdx.x;
  const int t = threadIdx.x;
  const float* bboxB = bbox + (size_t)b * NBOX * 4;
  const char*  scB   = (const char*)(scores_in + (size_t)b * NBOX);
  const uint32_t ldsBase = (uint32_t)(uintptr_t)(void*)s_sc;  // LDS byte offset

  // ---- stage scores[NBOX] into LDS ----
#if defined(__gfx1250__) && __has_builtin(__builtin_amdgcn_tensor_load_to_lds)
  if (t < 32) {                           // wave 0 issues one TDM descriptor
    uint64_t ga = (uint64_t)(uintptr_t)scB;
    v4u g0;
    g0[0] = 1u;                                               // count=1, user D#
    g0[1] = ldsBase;                                          // lds_addr (bytes)
    g0[2] = (uint32_t)ga;                                     // global_addr[31:0]
    g0[3] = (uint32_t)((ga >> 32) & 0x01FFFFFFu) | (2u << 30);// addr[56:32] | type=2
    v8i g1;
    g1[0] = (int)(2u << 16);                                  // data_size=4B, mask=0
    g1[1] = (int)(((uint32_t)NBOX & 0xFFFFu) << 16);          // tensor_dim0[15:0]
    g1[2] = (int)(((uint32_t)NBOX >> 16) | (1u << 16));       // dim0[31:16] | tensor_dim1=1
    g1[3] = (int)(((uint32_t)NBOX & 0xFFFFu) << 16);          // tile_dim0 = NBOX
    g1[4] = 0;                                                // tile_dim1/2 unused
    g1[5] = (int)NBOX;                                        // tensor_dim0_stride
    g1[6] = 0;
    g1[7] = 0;
    v4i z4 = {0, 0, 0, 0};
    v8i z8 = {0, 0, 0, 0, 0, 0, 0, 0};
    __builtin_amdgcn_tensor_load_to_lds(g0, g1, z4, z4, z8, 0);
    __builtin_amdgcn_s_wait_tensorcnt((short)0);
  }
#else
  // fallback: per-lane async global->LDS b128 DMA (ASYNCcnt-tracked)
#pragma unroll
  for (int k = 0; k < 7; ++k) {
    uint32_t off = (uint32_t)((k << 10) + t) * 16u;
    if (off < (uint32_t)SCORE_BYTES) {
      asm volatile("global_load_async_to_lds_b128 %0, %1, off"
                   :: "v"(ldsBase + off),
                      "v"((uint64_t)(uintptr_t)(scB + off))
                   : "memory");
    }
  }
  asm volatile("s_wait_asynccnt 0" ::: "memory");
#endif
  for (int j = NBOX + t; j < NPAD; j += BLOCK_NMS) s_sc[j] = NEG;  // pad tail
  __syncthreads();

  int vcount = MAX_B;

  for (int it = 0; it < MAX_B; ++it) {
    // ---- argmax over LDS scores: 7x ds_load_b128 per thread ----
    float bv = -3.4e38f;
    int   bi = 0;
#pragma unroll
    for (int k = 0; k < KGROUPS; ++k) {
      int j = (t << 2) + (k << 12);            // 4*t + 4096*k, 16B aligned
      v4f v = *(const v4f*)(s_sc + j);
      if (v.x > bv) { bv = v.x; bi = j + 0; }  // ascending index => first-on-tie
      if (v.y > bv) { bv = v.y; bi = j + 1; }
      if (v.z > bv) { bv = v.z; bi = j + 2; }
      if (v.w > bv) { bv = v.w; bi = j + 3; }
    }
    // wave32 reduction (value max, lower index wins ties)
#pragma unroll
    for (int off = 16; off > 0; off >>= 1) {
      float ov = __shfl_xor(bv, off, 32);
      int   oi = __shfl_xor(bi, off, 32);
      if (ov > bv || (ov == bv && oi < bi)) { bv = ov; bi = oi; }
    }
    int wave = t >> 5, lane = t & 31;
    if (lane == 0) { r_val[wave] = bv; r_idx[wave] = bi; }
    __syncthreads();
    if (wave == 0) {
      bv = r_val[lane];
      bi = r_idx[lane];
#pragma unroll
      for (int off = 16; off > 0; off >>= 1) {
        float ov = __shfl_xor(bv, off, 32);
        int   oi = __shfl_xor(bi, off, 32);
        if (ov > bv || (ov == bv && oi < bi)) { bv = ov; bi = oi; }
      }
      if (lane == 0) { bc[0] = bv; ((int*)bc)[1] = bi; }
    }
    __syncthreads();
    const float sv = bc[0];
    const int   si = ((int*)bc)[1];

    if (!(sv > SCORE_T)) {
      // Every remaining entry is NEG -> all later iterations invalid too.
      int    rem  = (MAX_B - it) * 6;
      float* base = outPred + ((size_t)b * MAX_B + it) * 6;
      for (int x = t; x < rem; x += BLOCK_NMS) base[x] = 0.0f;
      vcount = it;
      break;                                   // uniform across the block
    }

    // ---- selected box (uniform load) ----
    v4f sb = ((const v4f*)bboxB)[si];          // y1,x1,y2,x2
    float sArea = (sb.z - sb.x) * (sb.w - sb.y);

    if (t == 0) {
      float* o = outPred + ((size_t)b * MAX_B + it) * 6;
      o[0] = clamp01(sb.x);
      o[1] = clamp01(sb.y);
      o[2] = clamp01(sb.z);
      o[3] = clamp01(sb.w);
      o[4] = sv;
      o[5] = 0.0f;                             // single-class NMS -> class 0
    }

    // ---- IoU suppression: quad-wide, skip fully-dead quads ----
#pragma unroll
    for (int k = 0; k < KGROUPS; ++k) {
      int j = (t << 2) + (k << 12);
      if (j < NBOX) {                          // NBOX % 4 == 0 => quad fully in-range
        v4f s4 = *(const v4f*)(s_sc + j);
        float m = fmaxf(fmaxf(s4.x, s4.y), fmaxf(s4.z, s4.w));
        if (m != NEG) {                        // selected quad always passes (sv > T)
          const v4f* bp = (const v4f*)bboxB + j;   // 64B contiguous per thread
          v4f b0 = bp[0], b1 = bp[1], b2 = bp[2], b3 = bp[3];
          s4.x = supp_one(s4.x, j + 0, si, sb, sArea, b0);
          s4.y = supp_one(s4.y, j + 1, si, sb, sArea, b1);
          s4.z = supp_one(s4.z, j + 2, si, sb, sArea, b2);
          s4.w = supp_one(s4.w, j + 3, si, sb, sArea, b3);
          *(v4f*)(s_sc + j) = s4;
        }
      }
    }
    __syncthreads();
  }

  if (t == 0) outValid[b] = vcount;
}

// ---------------------------------------------------------------------------
extern "C" void kernel_launch(void* const* d_in, const int* in_sizes, int n_in,
                              void* d_out, int out_size, void* d_ws, size_t ws_size,
                              hipStream_t stream) {
  const float* bbox = (const float*)d_in[0];   // [32,25200,4]
  const float* p20  = (const float*)d_in[1];   // [32,25200,1]
  const float* c20  = (const float*)d_in[2];   // [32,25200,80]

  float* scores   = (float*)d_ws;                        // [32,25200] = 3.23 MB
  float* outPred  = (float*)d_out;                       // [32,100,6]
  int*   outValid = (int*)(outPred + BATCH * MAX_B * 6); // [32]

  const int total = BATCH * NBOX;                        // 806400
  score_kernel<<<(total + 255) / 256, 256, 0, stream>>>(p20, c20, scores, total);

  const size_t smemBytes = (size_t)NPAD * sizeof(float)  // scores in LDS
                         + 32 * sizeof(float)            // wave maxima
                         + 32 * sizeof(int)              // wave argmax indices
                         + 4 * sizeof(float);            // broadcast slot
  nms_kernel<<<BATCH, BLOCK_NMS, smemBytes, stream>>>(bbox, scores, outPred, outValid);
}